// TopkLrLinear_47304769798213
// MI455X (gfx1250) — compile-verified
//
#include <hip/hip_runtime.h>
#include <math.h>

typedef float v2f __attribute__((ext_vector_type(2)));
typedef float v8f __attribute__((ext_vector_type(8)));

#define NROWS 16384
#define KF    4096
#define OUTF  4096
#define RANKV 409
#define RPAD  416          // RANK padded to multiple of 32
#define NBLK  683
#define NPOOL 466489       // 683*683
#define TOPK  46649        // ceil(0.1 * 466489)
#define XP    36           // LDS row pitch (floats): bank-conflict-free, 16B-alignable

// ---------------- workspace layout (bytes) ----------------
static const size_t OFF_POOLED = 0;                       // 466489 f32
static const size_t OFF_HIST   = 1866240;                 // 2048 u32
static const size_t OFF_CTRL   = OFF_HIST + 8192;         // few ints
static const size_t OFF_G      = OFF_CTRL + 256;          // 16384 f32
static const size_t OFF_T      = OFF_G + 65536;           // 16384*416 f32

// ---------------- CDNA5 async global->LDS copy helpers ----------------
__device__ __forceinline__ unsigned __lds_off(const void* p) {
  // AS(3) pointers are offsets from the workgroup LDS base (what VDST expects)
  return (unsigned)(unsigned long long)(const __attribute__((address_space(3))) void*)p;
}
__device__ __forceinline__ void async_copy_b128(const void* gptr, void* lptr) {
  unsigned l = __lds_off(lptr);
  unsigned long long g = (unsigned long long)gptr;
  asm volatile("global_load_async_to_lds_b128 %0, %1, off"
               :: "v"(l), "v"(g) : "memory");
}
__device__ __forceinline__ void async_wait0() {
  asm volatile("s_wait_asynccnt 0" ::: "memory");
}

// ---------------- block-mean pooling of |W^T| ----------------
__global__ __launch_bounds__(256) void k_pool(const float* __restrict__ W,
                                              float* __restrict__ pooled) {
  int idx = blockIdx.x * 256 + threadIdx.x;
  if (idx >= NPOOL) return;
  int bx = idx / NBLK;             // block index along in-dim (i)
  int by = idx - bx * NBLK;        // block index along out-dim (o)
  int i0 = bx * 6, o0 = by * 6;
  int rc = (KF - i0 < 6) ? (KF - i0) : 6;
  int cc = (OUTF - o0 < 6) ? (OUTF - o0) : 6;
  float s = 0.f;
  for (int o = 0; o < cc; ++o) {
    const float* wp = W + (size_t)(o0 + o) * KF + i0;
    #pragma unroll
    for (int i = 0; i < 6; ++i)
      if (i < rc) s += fabsf(wp[i]);
  }
  pooled[idx] = s / (float)(rc * cc);
}

// ---------------- radix-select helpers ----------------
__global__ void k_init(unsigned* __restrict__ hist, int* __restrict__ ctrl) {
  if (threadIdx.x == 0) { ctrl[0] = TOPK; ctrl[1] = 0; ctrl[2] = 0; }
  for (int i = threadIdx.x; i < 2048; i += 256) hist[i] = 0;
}

__global__ __launch_bounds__(256) void k_hist(const float* __restrict__ pooled,
                                              unsigned* __restrict__ hist,
                                              const int* __restrict__ ctrl, int pass) {
  __shared__ unsigned lh[2048];
  int tid = threadIdx.x;
  for (int i = tid; i < 2048; i += 256) lh[i] = 0;
  __syncthreads();
  unsigned prefix = (pass > 0) ? (unsigned)ctrl[1] : 0u;
  int base = blockIdx.x * 2048;
  #pragma unroll
  for (int e = 0; e < 8; ++e) {
    int idx = base + tid + e * 256;
    if (idx < NPOOL) {
      unsigned bits = __float_as_uint(pooled[idx]);  // non-negative -> monotone
      if (pass == 0) {
        atomicAdd(&lh[bits >> 20], 1u);
      } else if (pass == 1) {
        if ((bits >> 20) == prefix) atomicAdd(&lh[(bits >> 10) & 1023u], 1u);
      } else {
        if ((bits >> 10) == prefix) atomicAdd(&lh[bits & 1023u], 1u);
      }
    }
  }
  __syncthreads();
  for (int i = tid; i < 2048; i += 256) {
    unsigned c = lh[i];
    if (c) atomicAdd(&hist[i], c);
  }
}

__global__ void k_pick(unsigned* __restrict__ hist, int* __restrict__ ctrl, int pass) {
  if (threadIdx.x == 0) {
    int nb = (pass == 0) ? 2048 : 1024;
    int rem = ctrl[0];
    long cum = 0;
    int sel = 0;
    for (int b = nb - 1; b >= 0; --b) {
      int c = (int)hist[b];
      if (cum + c >= rem) { sel = b; break; }
      cum += c;
    }
    ctrl[0] = rem - (int)cum;
    if (pass == 0)      ctrl[1] = sel;
    else if (pass == 1) ctrl[1] = (ctrl[1] << 10) | sel;
    else ((unsigned*)ctrl)[2] = (((unsigned)ctrl[1]) << 10) | (unsigned)sel;  // thr bits
  }
  __syncthreads();
  for (int i = threadIdx.x; i < 2048; i += 256) hist[i] = 0;
}

// ---------------- t = x @ lr1 (padded to 416 cols) + fused gate GEMV ----------------
__global__ __launch_bounds__(256) void k_lr1gate(
    const float* __restrict__ X, const float* __restrict__ LR1,
    const float* __restrict__ GW, const float* __restrict__ GB,
    float* __restrict__ T, float* __restrict__ G) {
  __shared__ __align__(16) float smA[128 * XP];  // x tile [row][k]
  __shared__ __align__(16) float smB[32 * 128];  // lr1 tile [k][r]
  __shared__ __align__(16) float smGW[32];

  const int tid = threadIdx.x;
  const int wid = tid >> 5;
  const int lane = tid & 31;
  const int h = lane >> 4;
  const int l = lane & 15;
  const int wm = wid >> 2;   // 0..1 -> 64-row slab
  const int wn = wid & 3;    // 0..3 -> 32-col slab
  const int m0 = blockIdx.x * 128;
  const int n0 = blockIdx.y * 128;
  const bool do_gate = (blockIdx.y == 0);

  v8f acc[4][2];
  #pragma unroll
  for (int mt = 0; mt < 4; ++mt)
    #pragma unroll
    for (int nt = 0; nt < 2; ++nt)
      acc[mt][nt] = (v8f){0.f, 0.f, 0.f, 0.f, 0.f, 0.f, 0.f, 0.f};

  float gacc = 0.f;

  for (int k0 = 0; k0 < KF; k0 += 32) {
    __syncthreads();
    #pragma unroll
    for (int e = 0; e < 4; ++e) {           // x tile: 128x32, async L2->LDS
      int f4 = tid + e * 256;
      int row = f4 >> 3;
      int c4 = (f4 & 7) << 2;
      async_copy_b128(X + (size_t)(m0 + row) * KF + k0 + c4, &smA[row * XP + c4]);
    }
    #pragma unroll
    for (int e = 0; e < 16; ++e) {          // lr1 tile [32][128], zero-pad r>=409
      int flat = tid + e * 256;
      int rr = flat >> 7;
      int cl = flat & 127;
      int r = n0 + cl;
      smB[rr * 128 + cl] = (r < RANKV) ? LR1[(size_t)(k0 + rr) * RANKV + r] : 0.f;
    }
    if (do_gate && tid < 32) smGW[tid] = GW[k0 + tid];
    async_wait0();
    __syncthreads();

    #pragma unroll
    for (int kk = 0; kk < 32; kk += 4) {
      v2f a[4], b[2];
      #pragma unroll
      for (int mt = 0; mt < 4; ++mt)
        a[mt] = *(const v2f*)(&smA[(wm * 64 + mt * 16 + l) * XP + kk + 2 * h]);
      #pragma unroll
      for (int nt = 0; nt < 2; ++nt) {
        int cn = wn * 32 + nt * 16 + l;
        v2f bb;
        bb.x = smB[(kk + 2 * h) * 128 + cn];
        bb.y = smB[(kk + 2 * h + 1) * 128 + cn];
        b[nt] = bb;
      }
      #pragma unroll
      for (int mt = 0; mt < 4; ++mt)
        #pragma unroll
        for (int nt = 0; nt < 2; ++nt)
          acc[mt][nt] = __builtin_amdgcn_wmma_f32_16x16x4_f32(
              false, a[mt], false, b[nt], (short)0, acc[mt][nt], false, false);
    }

    if (do_gate) {                          // per-wave 16 rows, 2 lanes/row
      int rrow = wid * 16 + l;
      int kb = h * 16;
      #pragma unroll
      for (int j = 0; j < 4; ++j) {
        float4 xv = *(const float4*)(&smA[rrow * XP + kb + j * 4]);
        float4 gv = *(const float4*)(&smGW[kb + j * 4]);
        gacc += xv.x * gv.x + xv.y * gv.y + xv.z * gv.z + xv.w * gv.w;
      }
    }
  }

  #pragma unroll
  for (int mt = 0; mt < 4; ++mt)
    #pragma unroll
    for (int nt = 0; nt < 2; ++nt) {
      int r = n0 + wn * 32 + nt * 16 + l;
      if (r < RPAD) {
        #pragma unroll
        for (int q = 0; q < 8; ++q) {
          int m = m0 + wm * 64 + mt * 16 + h * 8 + q;
          T[(size_t)m * RPAD + r] = acc[mt][nt][q];
        }
      }
    }

  if (do_gate) {
    gacc += __shfl_xor(gacc, 16, 32);
    if (h == 0) {
      int m = m0 + wid * 16 + l;
      G[m] = 1.f / (1.f + expf(-(gacc + GB[0])));
    }
  }
}

// ---------------- fused main: (g*x)@W'^T + ((1-g)*t)@lr2 + biases ----------------
__global__ __launch_bounds__(256) void k_main(
    const float* __restrict__ X, const float* __restrict__ W,
    const float* __restrict__ LR2,
    const float* __restrict__ SB, const float* __restrict__ LB,
    const float* __restrict__ pooled, const int* __restrict__ ctrl,
    const float* __restrict__ T, const float* __restrict__ G,
    float* __restrict__ OUT) {
  __shared__ __align__(16) float smA[128 * XP];  // x tile / t tile (unscaled)
  __shared__ __align__(16) float smB[128 * XP];  // masked W tile / lr2 tile [32][128]
  __shared__ float smG[128];
  __shared__ float smSB[128];
  __shared__ float smLB[128];

  const int tid = threadIdx.x;
  const int wid = tid >> 5;
  const int lane = tid & 31;
  const int h = lane >> 4;
  const int l = lane & 15;
  const int wm = wid >> 2;
  const int wn = wid & 3;
  const int m0 = blockIdx.x * 128;
  const int n0 = blockIdx.y * 128;

  if (tid < 128) {
    smG[tid]  = G[m0 + tid];
    smSB[tid] = SB[n0 + tid];
    smLB[tid] = LB[n0 + tid];
  }
  const float thr = __uint_as_float(((const unsigned*)ctrl)[2]);
  __syncthreads();

  // Per-lane row-gate scales for A-fragment scaling (row = wm*64+mt*16+l for both halves)
  float gm[4], hm[4];
  #pragma unroll
  for (int mt = 0; mt < 4; ++mt) {
    gm[mt] = smG[wm * 64 + mt * 16 + l];
    hm[mt] = 1.f - gm[mt];
  }

  v8f acc[4][2];
  #pragma unroll
  for (int mt = 0; mt < 4; ++mt)
    #pragma unroll
    for (int nt = 0; nt < 2; ++nt)
      acc[mt][nt] = (v8f){0.f, 0.f, 0.f, 0.f, 0.f, 0.f, 0.f, 0.f};

  // ---- sparse branch: K = 4096 ----
  for (int k0 = 0; k0 < KF; k0 += 32) {
    __syncthreads();
    #pragma unroll
    for (int e = 0; e < 4; ++e) {            // x tile: async L2->LDS (pure copy)
      int f4 = tid + e * 256;
      int row = f4 >> 3;
      int c4 = (f4 & 7) << 2;
      async_copy_b128(X + (size_t)(m0 + row) * KF + k0 + c4, &smA[row * XP + c4]);
    }
    #pragma unroll
    for (int e = 0; e < 4; ++e) {            // masked W (VALU; overlaps asyncs)
      int f4 = tid + e * 256;
      int row = f4 >> 3;
      int c4 = (f4 & 7) << 2;
      int o = n0 + row;
      int ob = o / 6;
      float4 v = *(const float4*)(W + (size_t)o * KF + k0 + c4);
      float vv[4] = {v.x, v.y, v.z, v.w};
      #pragma unroll
      for (int j = 0; j < 4; ++j) {
        int ii = k0 + c4 + j;
        float p = pooled[(ii / 6) * NBLK + ob];
        vv[j] = (p >= thr) ? vv[j] : 0.f;
      }
      v.x = vv[0]; v.y = vv[1]; v.z = vv[2]; v.w = vv[3];
      *(float4*)(&smB[row * XP + c4]) = v;
    }
    async_wait0();
    __syncthreads();

    #pragma unroll
    for (int kk = 0; kk < 32; kk += 4) {
      v2f a[4], b[2];
      #pragma unroll
      for (int mt = 0; mt < 4; ++mt)
        a[mt] = (*(const v2f*)(&smA[(wm * 64 + mt * 16 + l) * XP + kk + 2 * h])) * gm[mt];
      #pragma unroll
      for (int nt = 0; nt < 2; ++nt)
        b[nt] = *(const v2f*)(&smB[(wn * 32 + nt * 16 + l) * XP + kk + 2 * h]);
      #pragma unroll
      for (int mt = 0; mt < 4; ++mt)
        #pragma unroll
        for (int nt = 0; nt < 2; ++nt)
          acc[mt][nt] = __builtin_amdgcn_wmma_f32_16x16x4_f32(
              false, a[mt], false, b[nt], (short)0, acc[mt][nt], false, false);
    }
  }

  // ---- low-rank branch: R = 416 (padded) ----
  for (int r0 = 0; r0 < RPAD; r0 += 32) {
    __syncthreads();
    #pragma unroll
    for (int e = 0; e < 4; ++e) {            // t tile: async L2->LDS (pure copy)
      int f4 = tid + e * 256;
      int row = f4 >> 3;
      int c4 = (f4 & 7) << 2;
      async_copy_b128(T + (size_t)(m0 + row) * RPAD + r0 + c4, &smA[row * XP + c4]);
    }
    if (r0 < RPAD - 32) {                    // lr2 tile fully in-bounds: async b128
      #pragma unroll
      for (int e = 0; e < 4; ++e) {
        int f4 = tid + e * 256;              // 32x128 = 1024 float4
        int rr = f4 >> 5;
        int c4 = (f4 & 31) << 2;
        async_copy_b128(LR2 + (size_t)(r0 + rr) * OUTF + n0 + c4, &smB[rr * 128 + c4]);
      }
    } else {                                 // last chunk: zero-pad r>=409
      #pragma unroll
      for (int e = 0; e < 16; ++e) {
        int flat = tid + e * 256;
        int rr = flat >> 7;
        int cl = flat & 127;
        int r = r0 + rr;
        smB[rr * 128 + cl] = (r < RANKV) ? LR2[(size_t)r * OUTF + n0 + cl] : 0.f;
      }
    }
    async_wait0();
    __syncthreads();

    #pragma unroll
    for (int kk = 0; kk < 32; kk += 4) {
      v2f a[4], b[2];
      #pragma unroll
      for (int mt = 0; mt < 4; ++mt)
        a[mt] = (*(const v2f*)(&smA[(wm * 64 + mt * 16 + l) * XP + kk + 2 * h])) * hm[mt];
      #pragma unroll
      for (int nt = 0; nt < 2; ++nt) {
        int cn = wn * 32 + nt * 16 + l;
        v2f bb;
        bb.x = smB[(kk + 2 * h) * 128 + cn];
        bb.y = smB[(kk + 2 * h + 1) * 128 + cn];
        b[nt] = bb;
      }
      #pragma unroll
      for (int mt = 0; mt < 4; ++mt)
        #pragma unroll
        for (int nt = 0; nt < 2; ++nt)
          acc[mt][nt] = __builtin_amdgcn_wmma_f32_16x16x4_f32(
              false, a[mt], false, b[nt], (short)0, acc[mt][nt], false, false);
    }
  }

  // ---- epilogue: + g*sb + (1-g)*lb ----
  #pragma unroll
  for (int mt = 0; mt < 4; ++mt)
    #pragma unroll
    for (int nt = 0; nt < 2; ++nt) {
      int nl = wn * 32 + nt * 16 + l;
      float sb = smSB[nl];
      float lb = smLB[nl];
      int n = n0 + nl;
      #pragma unroll
      for (int q = 0; q < 8; ++q) {
        int ml = wm * 64 + mt * 16 + h * 8 + q;
        float gr = smG[ml];
        OUT[(size_t)(m0 + ml) * OUTF + n] = acc[mt][nt][q] + gr * sb + (1.f - gr) * lb;
      }
    }
}

// ---------------- host launcher ----------------
extern "C" void kernel_launch(void* const* d_in, const int* in_sizes, int n_in,
                              void* d_out, int out_size, void* d_ws, size_t ws_size,
                              hipStream_t stream) {
  const float* X   = (const float*)d_in[0];
  const float* GW  = (const float*)d_in[1];
  const float* GB  = (const float*)d_in[2];
  const float* W   = (const float*)d_in[3];
  const float* SB  = (const float*)d_in[4];
  const float* LR1 = (const float*)d_in[5];
  const float* LR2 = (const float*)d_in[6];
  const float* LB  = (const float*)d_in[7];
  float* OUT = (float*)d_out;

  char* ws = (char*)d_ws;
  float*    pooled = (float*)(ws + OFF_POOLED);
  unsigned* hist   = (unsigned*)(ws + OFF_HIST);
  int*      ctrl   = (int*)(ws + OFF_CTRL);
  float*    G      = (float*)(ws + OFF_G);
  float*    T      = (float*)(ws + OFF_T);

  // 1) block-mean pooling
  k_pool<<<(NPOOL + 255) / 256, 256, 0, stream>>>(W, pooled);
  // 2) 3-pass radix select of the TOPK-th largest pooled value
  k_init<<<1, 256, 0, stream>>>(hist, ctrl);
  int hgrid = (NPOOL + 2047) / 2048;
  k_hist<<<hgrid, 256, 0, stream>>>(pooled, hist, ctrl, 0);
  k_pick<<<1, 256, 0, stream>>>(hist, ctrl, 0);
  k_hist<<<hgrid, 256, 0, stream>>>(pooled, hist, ctrl, 1);
  k_pick<<<1, 256, 0, stream>>>(hist, ctrl, 1);
  k_hist<<<hgrid, 256, 0, stream>>>(pooled, hist, ctrl, 2);
  k_pick<<<1, 256, 0, stream>>>(hist, ctrl, 2);
  // 3) t = x @ lr1 (+zero pad) and gate
  k_lr1gate<<<dim3(NROWS / 128, 4), 256, 0, stream>>>(X, LR1, GW, GB, T, G);
  // 4) fused masked GEMM + low-rank + gated combine
  k_main<<<dim3(NROWS / 128, OUTF / 128), 256, 0, stream>>>(
      X, W, LR2, SB, LB, pooled, ctrl, T, G, OUT);
}